// ROIHeads_46291157516736
// MI455X (gfx1250) — compile-verified
//
#include <hip/hip_runtime.h>
#include <hip/hip_bf16.h>
#include <cstdint>

typedef __attribute__((ext_vector_type(2))) float v2f;
typedef __attribute__((ext_vector_type(8))) float v8f;

#define NIMG 4
#define NPROP 8192
#define NGT 128
#define NCLS 80
#define NLOG 81
#define CAP 131072
#define NMS_T 0.5f
#define SCLAMP 4.1351665567423560f

// ---------------------------------------------------------------- K0: init
__global__ void k_init(int* __restrict__ cnt) {
    if (threadIdx.x < NIMG) cnt[threadIdx.x] = 0;
}

// ---------------------------------------------------------------- K1: IoU match -> labels
// GT boxes staged into LDS with CDNA5 async global->LDS DMA.
__global__ void k_match(const float* __restrict__ gtb, const int* __restrict__ gtl,
                        const float* __restrict__ pb, int* __restrict__ labels) {
    int img = blockIdx.x;
    int n = blockIdx.y * 256 + threadIdx.x;
    __shared__ float4 s_gt[NGT];
    if (threadIdx.x < NGT) {
        const float4* gp = (const float4*)gtb + (size_t)img * NGT + threadIdx.x;
        unsigned lds = (unsigned)(size_t)(&s_gt[threadIdx.x]);
        unsigned long long ga = (unsigned long long)(size_t)gp;
        asm volatile("global_load_async_to_lds_b128 %0, %1, off" :: "v"(lds), "v"(ga) : "memory");
    }
    asm volatile("s_wait_asynccnt 0x0" ::: "memory");
    __syncthreads();

    float4 p = ((const float4*)pb)[(size_t)img * NPROP + n];
    float areaB = (p.z - p.x) * (p.w - p.y);
    float best = -1.0f; int bi = 0;
    for (int g = 0; g < NGT; ++g) {
        float4 gb = s_gt[g];
        float lx = fmaxf(gb.x, p.x), ly = fmaxf(gb.y, p.y);
        float rx = fminf(gb.z, p.z), ry = fminf(gb.w, p.w);
        float iw = fmaxf(rx - lx, 0.f), ih = fmaxf(ry - ly, 0.f);
        float inter = iw * ih;
        float areaA = (gb.z - gb.x) * (gb.w - gb.y);
        float iou = inter / (areaA + areaB - inter + 1e-9f);
        if (iou > best) { best = iou; bi = g; }   // strict > keeps first max (jnp.argmax)
    }
    int lab = (best >= 0.5f) ? gtl[img * NGT + bi] : 0;  // FG==BG==0.5 -> no -1 band
    labels[(size_t)img * NPROP + n] = lab;
}

// ---------------------------------------------------------------- K2: softmax (WMMA rowsum) + decode + compact candidates
// 256 thr = 8 waves, 16 proposals/wave, 128 proposals/block.
__global__ void k_cand(const float* __restrict__ dl, const float* __restrict__ lg,
                       const float* __restrict__ pb, int* __restrict__ cnt,
                       float* __restrict__ sc, float4* __restrict__ bx,
                       int* __restrict__ cls, int* __restrict__ fid) {
    int img = blockIdx.x;
    int nb = blockIdx.y * 128;
    int wave = threadIdx.x >> 5, lane = threadIdx.x & 31;
    int m = (lane < 16) ? lane : (lane - 16);
    int koff = (lane < 16) ? 0 : 2;

    // --- WMMA row-sum of exps: D = A(16x4) * ones(4x16) + C, 21 chunks cover k=0..83
    size_t rowBase = ((size_t)(img * NPROP + nb + wave * 16 + m)) * NLOG;
    v8f acc = {};
    v2f ones; ones[0] = 1.0f; ones[1] = 1.0f;
    for (int ch = 0; ch < 21; ++ch) {
        int k0 = ch * 4 + koff;
        float e0 = (k0 < NLOG) ? expf(lg[rowBase + k0]) : 0.0f;
        float e1 = (k0 + 1 < NLOG) ? expf(lg[rowBase + k0 + 1]) : 0.0f;
        v2f a; a[0] = e0; a[1] = e1;
        acc = __builtin_amdgcn_wmma_f32_16x16x4_f32(false, a, false, ones,
                                                    (short)0, acc, false, false);
    }
    // rows 0..7 live in lane 0 (N=0), rows 8..15 in lane 16 (N=0)
    __shared__ float s_sum[128];
    if (lane == 0)  { for (int r = 0; r < 8; ++r) s_sum[wave * 16 + r] = acc[r]; }
    if (lane == 16) { for (int r = 0; r < 8; ++r) s_sum[wave * 16 + 8 + r] = acc[r]; }
    __syncthreads();

    // --- candidate scan: 2 threads per proposal, 40 classes each
    int q = threadIdx.x >> 1, h = threadIdx.x & 1;
    int n = nb + q;
    float S = s_sum[q];
    float thr = logf(0.05f * S);                 // prob > 0.05  <=>  logit > log(0.05*S)
    size_t gidx = (size_t)img * NPROP + n;
    float4 pbv = ((const float4*)pb)[gidx];
    float w = pbv.z - pbv.x, hh = pbv.w - pbv.y;
    float cx = pbv.x + 0.5f * w, cy = pbv.y + 0.5f * hh;
    const float* lrow = lg + gidx * NLOG;
    const float4* drow = (const float4*)(dl + gidx * (NCLS * 4));
    for (int c = h * 40; c < h * 40 + 40; ++c) {
        float l = lrow[1 + c];
        if (l > thr) {
            float4 d = drow[c];
            float dx = d.x * 0.1f, dy = d.y * 0.1f, dw = d.z * 0.2f, dh = d.w * 0.2f;
            float pcx = dx * w + cx, pcy = dy * hh + cy;
            float pw = expf(fminf(dw, SCLAMP)) * w;
            float ph = expf(fminf(dh, SCLAMP)) * hh;
            float x1 = fminf(fmaxf(pcx - 0.5f * pw, 0.f), 1024.f);
            float y1 = fminf(fmaxf(pcy - 0.5f * ph, 0.f), 1024.f);
            float x2 = fminf(fmaxf(pcx + 0.5f * pw, 0.f), 1024.f);
            float y2 = fminf(fmaxf(pcy + 0.5f * ph, 0.f), 1024.f);
            float score = expf(l) / S;
            int slot = atomicAdd(&cnt[img], 1);
            if (slot < CAP) {
                size_t s = (size_t)img * CAP + slot;
                sc[s] = score;
                bx[s] = make_float4(x1, y1, x2, y2);
                cls[s] = c;
                fid[s] = n * NCLS + c;           // reference flat id; argmax tie-break
            }
        }
    }
}

// ---------------------------------------------------------------- K3: greedy NMS, 1 block/image
__global__ void __launch_bounds__(1024) k_nms(const int* __restrict__ cnt,
                                              float* __restrict__ sc_, const float4* __restrict__ bx_,
                                              const int* __restrict__ cls_, const int* __restrict__ fid_,
                                              float* __restrict__ out) {
    int img = blockIdx.x, tid = threadIdx.x;
    float* sc = sc_ + (size_t)img * CAP;
    const float4* bx = bx_ + (size_t)img * CAP;
    const int* cls = cls_ + (size_t)img * CAP;
    const int* fid = fid_ + (size_t)img * CAP;
    int M = cnt[img]; if (M > CAP) M = CAP;

    __shared__ float s_v[1024]; __shared__ int s_f[1024]; __shared__ int s_s[1024];
    __shared__ float4 s_box; __shared__ int s_cls; __shared__ int s_keep;

    for (int it = 0; it < 100; ++it) {
        float bv = -1e30f; int bf = 0x7fffffff; int bs = -1;
        for (int i = tid; i < M; i += 1024) {
            float v = sc[i]; int f = fid[i];
            if (v > bv || (v == bv && f < bf)) { bv = v; bf = f; bs = i; }
        }
        s_v[tid] = bv; s_f[tid] = bf; s_s[tid] = bs;
        __syncthreads();
        for (int st = 512; st > 0; st >>= 1) {
            if (tid < st) {
                float v2 = s_v[tid + st]; int f2 = s_f[tid + st];
                if (v2 > s_v[tid] || (v2 == s_v[tid] && f2 < s_f[tid])) {
                    s_v[tid] = v2; s_f[tid] = f2; s_s[tid] = s_s[tid + st];
                }
            }
            __syncthreads();
        }
        if (tid == 0) {
            float v = s_v[0]; int slot = s_s[0];
            int keep = (slot >= 0) && (v > 0.0f);
            float4 ob = make_float4(0.f, 0.f, 0.f, 0.f); float os = 0.f; int oc = 0;
            if (keep) { ob = bx[slot]; os = v; oc = cls[slot] + 1; s_box = ob; s_cls = cls[slot]; }
            if (slot >= 0) sc[slot] = -1.0f;
            s_keep = keep;
            out[img * 400 + it * 4 + 0] = ob.x;
            out[img * 400 + it * 4 + 1] = ob.y;
            out[img * 400 + it * 4 + 2] = ob.z;
            out[img * 400 + it * 4 + 3] = ob.w;
            out[1600 + img * 100 + it] = os;
            out[2000 + img * 100 + it] = (float)oc;
        }
        __threadfence_block();
        __syncthreads();
        if (s_keep) {
            float4 sb = s_box; int scl = s_cls;
            float aa = (sb.z - sb.x) * (sb.w - sb.y);
            for (int i = tid; i < M; i += 1024) {
                if (cls[i] != scl) continue;     // CLS_OFFSET separation => cross-class IoU == 0
                float4 b = bx[i];
                float lx = fmaxf(sb.x, b.x), ly = fmaxf(sb.y, b.y);
                float rx = fminf(sb.z, b.z), ry = fminf(sb.w, b.w);
                float iw = fmaxf(rx - lx, 0.f), ih = fmaxf(ry - ly, 0.f);
                float inter = iw * ih;
                float ab = (b.z - b.x) * (b.w - b.y);
                float iou = inter / (aa + ab - inter + 1e-9f);
                if (iou > NMS_T) sc[i] = -1.0f;
            }
        }
        __threadfence_block();
        __syncthreads();
    }
}

// ---------------------------------------------------------------- K4: pos/neg subsample + top_k ordering
__device__ __forceinline__ int blockReduceSum(int v, int* s_red, int lane, int wave) {
    for (int o = 16; o > 0; o >>= 1) v += __shfl_down(v, o);
    if (lane == 0) s_red[wave] = v;
    __syncthreads();
    if (wave == 0) {
        int x = s_red[lane];
        for (int o = 16; o > 0; o >>= 1) x += __shfl_down(x, o);
        if (lane == 0) s_red[0] = x;
    }
    __syncthreads();
    int r = s_red[0];
    __syncthreads();
    return r;
}

__global__ void __launch_bounds__(1024) k_sample(const float* __restrict__ rnd,
                                                 const int* __restrict__ labels,
                                                 float* __restrict__ out) {
    int img = blockIdx.x, tid = threadIdx.x;
    int lane = tid & 31, wave = tid >> 5;
    __shared__ float s_rand[NPROP];
    __shared__ unsigned char s_pos[NPROP];
    __shared__ unsigned long long s_key[512];
    __shared__ int s_red[32];
    __shared__ int s_cnt;

    for (int i = tid; i < NPROP; i += 1024) {
        s_rand[i] = rnd[(size_t)img * NPROP + i];
        s_pos[i] = labels[(size_t)img * NPROP + i] > 0;
    }
    if (tid == 0) s_cnt = 0;
    __syncthreads();

    int c = 0;
    for (int i = tid; i < NPROP; i += 1024) c += s_pos[i];
    int tp = blockReduceSum(c, s_red, lane, wave);
    int np = tp < 128 ? tp : 128;
    int negc = NPROP - tp;
    int nn = (512 - np) < negc ? (512 - np) : negc;

    // 45-bit key: rand desc, tie -> lower index first (matches stable argsort/top_k)
    auto key45 = [&](int i) -> unsigned long long {
        return ((unsigned long long)__float_as_uint(s_rand[i]) << 13) |
               (unsigned long long)(NPROP - 1 - i);
    };
    // k-th largest key within group (bitwise search); k==gcount -> 0 (select all)
    auto groupKth = [&](int wantPos, int k, int gcount) -> unsigned long long {
        if (k >= gcount) return 0ULL;
        unsigned long long T = 0ULL;
        for (int b = 44; b >= 0; --b) {
            unsigned long long trial = T | (1ULL << b);
            int cc = 0;
            for (int i = tid; i < NPROP; i += 1024)
                if (((int)s_pos[i]) == wantPos && key45(i) >= trial) cc++;
            if (blockReduceSum(cc, s_red, lane, wave) >= k) T = trial;
        }
        return T;
    };
    unsigned long long Tpos = groupKth(1, np, tp);
    unsigned long long Tneg = groupKth(0, nn, negc);

    auto key46 = [&](int i) -> unsigned long long {
        unsigned long long k = key45(i);
        unsigned long long sel = s_pos[i] ? (k >= Tpos) : (k >= Tneg);
        return k | (sel << 45);                  // prio = rand + 2*sel, same ordering
    };
    // 512-th largest combined key
    unsigned long long T512 = 0ULL;
    for (int b = 45; b >= 0; --b) {
        unsigned long long trial = T512 | (1ULL << b);
        int cc = 0;
        for (int i = tid; i < NPROP; i += 1024)
            if (key46(i) >= trial) cc++;
        if (blockReduceSum(cc, s_red, lane, wave) >= 512) T512 = trial;
    }
    // compact exactly 512 (keys distinct via index bits)
    for (int i = tid; i < NPROP; i += 1024) {
        unsigned long long k = key46(i);
        if (k >= T512) {
            int p = atomicAdd(&s_cnt, 1);
            if (p < 512) s_key[p] = k;
        }
    }
    __syncthreads();
    // bitonic sort 512 keys, descending
    for (int kk = 2; kk <= 512; kk <<= 1) {
        for (int j = kk >> 1; j > 0; j >>= 1) {
            __syncthreads();
            if (tid < 512) {
                int ixj = tid ^ j;
                if (ixj > tid) {
                    unsigned long long a = s_key[tid], b = s_key[ixj];
                    bool up = (tid & kk) == 0;
                    if (up ? (a < b) : (a > b)) { s_key[tid] = b; s_key[ixj] = a; }
                }
            }
        }
    }
    __syncthreads();
    if (tid < 512) {
        unsigned long long k = s_key[tid];
        int n = NPROP - 1 - (int)(k & 0x1FFFULL);
        out[2400 + img * 512 + tid] = (float)n;
        out[4448 + img * 512 + tid] = (float)labels[(size_t)img * NPROP + n];
    }
}

// ---------------------------------------------------------------- launch
extern "C" void kernel_launch(void* const* d_in, const int* in_sizes, int n_in,
                              void* d_out, int out_size, void* d_ws, size_t ws_size,
                              hipStream_t stream) {
    const float* gtb = (const float*)d_in[0];
    const int*   gtl = (const int*)d_in[1];
    const float* pb  = (const float*)d_in[2];
    const float* dl  = (const float*)d_in[3];
    const float* lg  = (const float*)d_in[4];
    const float* rnd = (const float*)d_in[5];
    float* out = (float*)d_out;
    char* ws = (char*)d_ws;

    auto al = [](size_t x) { return (x + 255) & ~(size_t)255; };
    size_t o = 0;
    int*    cnt    = (int*)(ws + o);    o = al(o + NIMG * sizeof(int));
    int*    labels = (int*)(ws + o);    o = al(o + (size_t)NIMG * NPROP * sizeof(int));
    float*  sc     = (float*)(ws + o);  o = al(o + (size_t)NIMG * CAP * sizeof(float));
    float4* bx     = (float4*)(ws + o); o = al(o + (size_t)NIMG * CAP * sizeof(float4));
    int*    cls    = (int*)(ws + o);    o = al(o + (size_t)NIMG * CAP * sizeof(int));
    int*    fid    = (int*)(ws + o);    o = al(o + (size_t)NIMG * CAP * sizeof(int));
    (void)in_sizes; (void)n_in; (void)out_size; (void)ws_size;

    k_init<<<1, 64, 0, stream>>>(cnt);
    k_match<<<dim3(NIMG, NPROP / 256), 256, 0, stream>>>(gtb, gtl, pb, labels);
    k_cand<<<dim3(NIMG, NPROP / 128), 256, 0, stream>>>(dl, lg, pb, cnt, sc, bx, cls, fid);
    k_nms<<<NIMG, 1024, 0, stream>>>(cnt, sc, bx, cls, fid, out);
    k_sample<<<NIMG, 1024, 0, stream>>>(rnd, labels, out);
}